// DSoftKI_56684978372611
// MI455X (gfx1250) — compile-verified
//
#include <hip/hip_runtime.h>

#define BB 16384
#define MM 512
#define DD 8

typedef float v2f __attribute__((ext_vector_type(2)));
typedef float v8f __attribute__((ext_vector_type(8)));

// workspace layout (floats):
//   [0, 20*512)             Bfeat[k][m]: K=20-padded feature matrix for dist^2 GEMM
//   [20*512, 20*512+512*16) B2[m][n]:  n<8 -> 1/T^2,  n>=8 -> z/T
#define WS_BFEAT 0
#define WS_BM2   (20 * MM)

__global__ void dsoftki_prep(const float* __restrict__ z,
                             const float* __restrict__ T,
                             float* __restrict__ ws) {
  int m = blockIdx.x * blockDim.x + threadIdx.x;
  if (m >= MM) return;
  float sumz2 = 0.f;
  float it2[DD], zt[DD];
#pragma unroll
  for (int d = 0; d < DD; ++d) {
    float zv = z[m * DD + d];
    float tv = T[m * DD + d];
    float it = __builtin_amdgcn_rcpf(tv);
    it2[d] = it * it;
    zt[d]  = zv * it;
    sumz2 += zv * zv;
  }
  float* Bf = ws + WS_BFEAT;
#pragma unroll
  for (int d = 0; d < DD; ++d) {
    Bf[d * MM + m]       = it2[d];        // pairs with x_d^2
    Bf[(8 + d) * MM + m] = -2.0f * zt[d]; // pairs with x_d
  }
  Bf[16 * MM + m] = sumz2;                // pairs with 1
  Bf[17 * MM + m] = 0.f;
  Bf[18 * MM + m] = 0.f;
  Bf[19 * MM + m] = 0.f;
  float* B2 = ws + WS_BM2;
#pragma unroll
  for (int d = 0; d < DD; ++d) {
    B2[m * 16 + d]     = it2[d];
    B2[m * 16 + 8 + d] = zt[d];
  }
}

__launch_bounds__(32)
__global__ void dsoftki_main(const float* __restrict__ x,
                             const float* __restrict__ ws,
                             float* __restrict__ out) {
  __shared__ __align__(16) float xs[16 * DD]; // x tile (16 rows x 8)
  __shared__ float sd[16 * MM];               // dist[16][512]
  __shared__ float raw[16 * 16];              // mean-GEMM raw output
  __shared__ float dmin_s[16], isum_s[16];

  const int lane  = threadIdx.x;
  const int b0    = blockIdx.x * 16;
  const int row   = lane & 15;                // A/C row identity
  const int nidx  = lane & 15;                // B/C col identity
  const int koff  = (lane < 16) ? 0 : 2;      // A/B K sub-offset per lane half
  const int rbase = (lane < 16) ? 0 : 8;      // C/D row base per lane half
  const float* Bf = ws + WS_BFEAT;
  const float* B2 = ws + WS_BM2;

  // ---- load x tile (128 floats, 16B per lane) ----
  ((float4*)xs)[lane] = ((const float4*)(x + (size_t)b0 * DD))[lane];
  __syncthreads();

  // ---- per-lane A fragments: features [x^2(0..7) | x(8..15) | 1 | 0 0 0] ----
  float fa0[5], fa1[5];
#pragma unroll
  for (int c = 0; c < 5; ++c) {
    int j0 = 4 * c + koff;
    int j1 = j0 + 1;
    float x0 = xs[row * DD + (j0 & 7)];
    float x1 = xs[row * DD + (j1 & 7)];
    fa0[c] = (j0 < 8) ? x0 * x0 : (j0 < 16 ? x0 : (j0 == 16 ? 1.f : 0.f));
    fa1[c] = (j1 < 8) ? x1 * x1 : (j1 < 16 ? x1 : (j1 == 16 ? 1.f : 0.f));
  }

  // ---- dist^2 GEMM: [16x20] x [20x16] per m-tile, 32 tiles ----
  for (int mt = 0; mt < 32; ++mt) {
    const int col = mt * 16 + nidx;
    v8f acc = {};
#pragma unroll
    for (int c = 0; c < 5; ++c) {
      int k0 = 4 * c + koff;
      v2f a, b;
      a.x = fa0[c];
      a.y = fa1[c];
      b.x = Bf[k0 * MM + col];
      b.y = Bf[(k0 + 1) * MM + col];
      acc = __builtin_amdgcn_wmma_f32_16x16x4_f32(false, a, false, b,
                                                  (short)0, acc, false, false);
    }
#pragma unroll
    for (int v = 0; v < 8; ++v) {
      float dist = __builtin_amdgcn_sqrtf(fmaxf(acc[v], 0.f));
      sd[(rbase + v) * MM + col] = dist;
    }
  }
  __syncthreads();

  // ---- softmax stats per row: max(-dist) = -min(dist), and sum of exp ----
  {
    const int srow = lane >> 1;
    const int sub  = lane & 1;
    float dmin = 3.4e38f;
    for (int m = sub; m < MM; m += 2) dmin = fminf(dmin, sd[srow * MM + m]);
    dmin = fminf(dmin, __shfl_xor(dmin, 1, 32));
    float sum = 0.f;
    for (int m = sub; m < MM; m += 2) sum += __expf(dmin - sd[srow * MM + m]);
    sum += __shfl_xor(sum, 1, 32);
    if (sub == 0) {
      dmin_s[srow] = dmin;
      isum_s[srow] = __builtin_amdgcn_rcpf(sum);
    }
  }
  __syncthreads();

  // ---- mean_dd GEMM: S[16x512] x B2[512x16], S = W/(dist+eps) ----
  {
    const float dminr = dmin_s[row];
    const float isumr = isum_s[row];
    v8f macc = {};
    for (int kc = 0; kc < MM; kc += 4) {
      int k0 = kc + koff;
      float d0 = sd[row * MM + k0];
      float d1 = sd[row * MM + k0 + 1];
      v2f a, b;
      a.x = __expf(dminr - d0) * isumr * __builtin_amdgcn_rcpf(d0 + 1e-6f);
      a.y = __expf(dminr - d1) * isumr * __builtin_amdgcn_rcpf(d1 + 1e-6f);
      b.x = B2[k0 * 16 + nidx];
      b.y = B2[(k0 + 1) * 16 + nidx];
      macc = __builtin_amdgcn_wmma_f32_16x16x4_f32(false, a, false, b,
                                                   (short)0, macc, false, false);
    }
#pragma unroll
    for (int v = 0; v < 8; ++v) raw[(rbase + v) * 16 + nidx] = macc[v];
  }
  __syncthreads();

  // ---- output pass: W and deriv, coalesced in m, non-temporal (302MB stream) ----
  for (int mt = 0; mt < 16; ++mt) {
    const int m = mt * 32 + lane;
    float it2[DD], zt[DD];
#pragma unroll
    for (int d = 0; d < DD; ++d) {
      it2[d] = B2[m * 16 + d];
      zt[d]  = B2[m * 16 + 8 + d];
    }
    for (int r = 0; r < 16; ++r) {
      const float dminr = dmin_s[r];
      const float isumr = isum_s[r];
      float dist = sd[r * MM + m];
      float w = __expf(dminr - dist) * isumr;
      float s = w * __builtin_amdgcn_rcpf(dist + 1e-6f);
      __builtin_nontemporal_store(w, out + (size_t)(b0 + r) * MM + m);
#pragma unroll
      for (int d = 0; d < DD; ++d) {
        float xv = xs[r * DD + d];
        float md = xv * raw[r * 16 + d] - raw[r * 16 + 8 + d]; // mean_dd[b][d]
        // deriv = -W*(dd - mean_dd) = W*mean_dd - S*(x*invT2 - z/T)
        float od = w * md - s * (xv * it2[d] - zt[d]);
        __builtin_nontemporal_store(od, out + (size_t)(BB + (b0 + r) * DD + d) * MM + m);
      }
    }
  }
}

extern "C" void kernel_launch(void* const* d_in, const int* in_sizes, int n_in,
                              void* d_out, int out_size, void* d_ws, size_t ws_size,
                              hipStream_t stream) {
  const float* x = (const float*)d_in[0];
  const float* z = (const float*)d_in[1];
  const float* T = (const float*)d_in[2];
  float* out = (float*)d_out;
  float* ws  = (float*)d_ws;
  (void)in_sizes; (void)n_in; (void)out_size; (void)ws_size;

  hipLaunchKernelGGL(dsoftki_prep, dim3(2), dim3(256), 0, stream, z, T, ws);
  hipLaunchKernelGGL(dsoftki_main, dim3(BB / 16), dim3(32), 0, stream, x, ws, out);
}